// channel_clustering_hpro_loss_1958505087267
// MI455X (gfx1250) — compile-verified
//
#include <hip/hip_runtime.h>
#include <math.h>

#define BSZ   8
#define C_CH  4096
#define SPAT  891
#define DLEN  7128            // SPAT * BSZ
#define TK    64
#define NCHUNK 112            // ceil(7128/64), last chunk zero-padded
#define NROWS 4096            // total zs rows per level

typedef float v2f __attribute__((ext_vector_type(2)));
typedef float v8f __attribute__((ext_vector_type(8)));

// ---------------------------------------------------------------------------
// K1: build per-level cluster membership (serial per level; 4096 iters, cheap)
// grid = 2 (level), block = 32 (only thread 0 works)
// ---------------------------------------------------------------------------
__global__ void k_build(const int* __restrict__ im2c0,
                        const int* __restrict__ im2c1,
                        int* __restrict__ members,   // [2][4096] channel ids, cluster-major
                        int* __restrict__ rowstart,  // [2][64]
                        int* __restrict__ count,     // [2][64]
                        int* __restrict__ rowclu)    // [2][4096] global-row -> cluster
{
    int L = blockIdx.x;
    if (threadIdx.x != 0) return;
    const int* im2c = (L == 0) ? im2c0 : im2c1;
    int K = (L == 0) ? 64 : 16;
    int* cnt = count    + L * 64;
    int* rs  = rowstart + L * 64;
    int* mem = members  + L * C_CH;
    int* rc  = rowclu   + L * C_CH;
    for (int k = 0; k < 64; ++k) cnt[k] = 0;
    for (int ch = 0; ch < C_CH; ++ch) cnt[im2c[ch]]++;
    int acc = 0;
    for (int k = 0; k < K; ++k) { rs[k] = acc; acc += cnt[k]; }
    for (int k = K; k < 64; ++k) rs[k] = acc;
    int fill[64];
    for (int k = 0; k < K; ++k) fill[k] = rs[k];
    for (int ch = 0; ch < C_CH; ++ch) {
        int c = im2c[ch];
        mem[fill[c]] = ch;
        rc[fill[c]]  = c;
        fill[c]++;
    }
}

// ---------------------------------------------------------------------------
// K2: per (level, global-row, slab jj in 0..7) base offset into conv:
//     zs_val(gr, d) = conv[ jbase[gr*8 + d/891] + d%891 ]
// grid = 256, block = 256  (2*4096*8 = 65536 entries)
// ---------------------------------------------------------------------------
__global__ void k_jbase(const int* __restrict__ members,
                        const int* __restrict__ rowstart,
                        const int* __restrict__ count,
                        const int* __restrict__ rowclu,
                        int* __restrict__ jbase)     // [2][4096][8]
{
    int idx = blockIdx.x * blockDim.x + threadIdx.x;   // < 65536
    int L   = idx >> 15;
    int rem = idx & 32767;
    int gr  = rem >> 3;
    int jj  = rem & 7;
    int c   = rowclu[L * C_CH + gr];
    int rs  = rowstart[L * 64 + c];
    int nc  = count[L * 64 + c];
    int r   = gr - rs;
    int j   = 8 * r + jj;          // j in [0, 8*nc)
    int b   = j / nc;              // batch index in [0,8)
    int i   = j - b * nc;          // member index
    int ch  = members[L * C_CH + rs + i];
    jbase[idx] = (b * C_CH + ch) * SPAT;
}

// ---------------------------------------------------------------------------
// K3: WMMA GEMM  P[4096 x Kc] = A (scrambled rows) * B^T (cluster row-0s)
//     + per-row squared norms. One block per 16-row tile, 4 wave32s.
// ---------------------------------------------------------------------------
__global__ __launch_bounds__(128)
void k_gemm(const float* __restrict__ conv,
            const int*   __restrict__ jbaseL,     // [4096][8]
            const int*   __restrict__ rowstartL,  // [64]
            float* __restrict__ P,                // [4096][Kc]
            float* __restrict__ rnL,              // [4096]
            int Kc)
{
    __shared__ float As[16][TK + 1];
    __shared__ float Bs[TK][64 + 1];

    const int t      = threadIdx.x;
    const int rt     = blockIdx.x;       // row tile, 0..255
    const int dloc   = t & 63;
    const int rowoff = t >> 6;           // 0 or 1
    const int lane   = t & 31;
    const int w      = t >> 5;           // wave id 0..3
    const int half   = lane >> 4;        // 0: lanes 0-15, 1: lanes 16-31
    const int nn     = lane & 15;
    const int numCT  = Kc >> 4;          // 4 (K=64) or 1 (K=16)
    const int ct     = w;                // this wave's 16-col tile

    float accs[8] = {0.f,0.f,0.f,0.f,0.f,0.f,0.f,0.f};
    v8f acc = {};

    for (int chunk = 0; chunk < NCHUNK; ++chunk) {
        const int d  = chunk * TK + dloc;
        const bool ok = (d < DLEN);
        int jj = 0, s = 0;
        if (ok) { jj = d / SPAT; s = d - jj * SPAT; }

        // ---- gather A tile: 16 rows x 64 cols; thread covers 8 rows at fixed dloc
        #pragma unroll
        for (int ri = 0; ri < 8; ++ri) {
            const int mi = rowoff + 2 * ri;
            const int gr = rt * 16 + mi;
            float v = 0.f;
            if (ok) v = conv[jbaseL[gr * 8 + jj] + s];
            As[mi][dloc] = v;
            accs[ri] += v * v;           // row-norm partial (padding adds 0)
        }
        // ---- gather B tile: 64 (k) x Kc (clusters), from each cluster's row 0
        for (int n = rowoff; n < Kc; n += 2) {
            float v = 0.f;
            if (ok) {
                const int g0 = rowstartL[n];
                v = conv[jbaseL[g0 * 8 + jj] + s];
            }
            Bs[dloc][n] = v;
        }
        __syncthreads();

        // ---- MMA: 16 x V_WMMA_F32_16X16X4_F32 per chunk per active wave
        if (w < numCT) {
            #pragma unroll
            for (int kk = 0; kk < TK; kk += 4) {
                const int kb = kk + 2 * half;
                v2f a, b;
                a.x = As[nn][kb];                 // A: M=nn, K=kb
                a.y = As[nn][kb + 1];             //    M=nn, K=kb+1
                b.x = Bs[kb][ct * 16 + nn];       // B: K=kb,   N=nn
                b.y = Bs[kb + 1][ct * 16 + nn];   //    K=kb+1, N=nn
                acc = __builtin_amdgcn_wmma_f32_16x16x4_f32(
                        false, a, false, b, (short)0, acc, false, false);
            }
        }
        __syncthreads();
    }

    // ---- write P tile (D layout: VGPR v -> M = v + 8*half, N = nn)
    if (w < numCT) {
        #pragma unroll
        for (int v = 0; v < 8; ++v) {
            const int gr = rt * 16 + v + 8 * half;
            P[gr * Kc + ct * 16 + nn] = acc[v];
        }
    }

    // ---- deterministic row-norm reduction (reuse As; no float atomics)
    #pragma unroll
    for (int ri = 0; ri < 8; ++ri)
        As[rowoff + 2 * ri][dloc] = accs[ri];
    __syncthreads();
    if (t < 16) {
        float ssum = 0.f;
        for (int dl = 0; dl < TK; ++dl) ssum += As[t][dl];
        rnL[rt * 16 + t] = ssum;
    }
}

// ---------------------------------------------------------------------------
// K4: per-cluster conc + loss.  grid = 80 (64 level-0 clusters, 16 level-1)
// ---------------------------------------------------------------------------
__global__ __launch_bounds__(256)
void k_cluster(const int*   __restrict__ rowstart,
               const int*   __restrict__ count,
               const float* __restrict__ P0,
               const float* __restrict__ P1,
               const float* __restrict__ rn,      // [2][4096]
               float* __restrict__ concv,         // [2][64]
               float* __restrict__ lossc)         // [2][64]
{
    __shared__ float red[256];
    __shared__ float s_conc;

    const int t = threadIdx.x;
    const int L = (blockIdx.x < 64) ? 0 : 1;
    const int c = (L == 0) ? blockIdx.x : (blockIdx.x - 64);
    const int Kc = (L == 0) ? 64 : 16;
    const float* Pm  = (L == 0) ? P0 : P1;
    const float* rnL = rn + L * NROWS;
    const int nc = count[L * 64 + c];
    const int g0 = rowstart[L * 64 + c];

    // phase 1: conc = sum_r ||zs[r]-zs[0]|| / (nc * ln(nc+10))
    float part = 0.f;
    for (int r = t; r < nc; r += 256) {
        const int gr = g0 + r;
        const float sq = rnL[gr] - 2.f * Pm[gr * Kc + c] + rnL[g0];
        part += (sq > 0.f) ? sqrtf(sq) : 0.f;
    }
    red[t] = part;
    __syncthreads();
    for (int off = 128; off > 0; off >>= 1) {
        if (t < off) red[t] += red[t + off];
        __syncthreads();
    }
    if (t == 0) {
        s_conc = red[0] / ((float)nc * logf((float)nc + 10.f));
        concv[L * 64 + c] = s_conc;
    }
    __syncthreads();
    const float conc = s_conc;

    // phase 2: mean_r [ logsumexp_k(P[r,k]/conc) - P[r,c]/conc ]
    float lp = 0.f;
    if (nc > 1) {
        const float inv = 1.f / conc;
        for (int r = t; r < nc; r += 256) {
            const int gr = g0 + r;
            const float* row = Pm + gr * Kc;
            float mx = -3.4e38f;
            for (int k = 0; k < Kc; ++k) {
                const float q = row[k] * inv;
                mx = (q > mx) ? q : mx;
            }
            float se = 0.f;
            for (int k = 0; k < Kc; ++k)
                se += expf(row[k] * inv - mx);
            lp += (mx + logf(se)) - row[c] * inv;
        }
    }
    red[t] = lp;
    __syncthreads();
    for (int off = 128; off > 0; off >>= 1) {
        if (t < off) red[t] += red[t + off];
        __syncthreads();
    }
    if (t == 0)
        lossc[L * 64 + c] = (nc > 1) ? (red[0] / (float)nc) : 0.f;
}

// ---------------------------------------------------------------------------
// K5: final deterministic scalar reduction -> out[2]
// ---------------------------------------------------------------------------
__global__ void k_final(const float* __restrict__ lossc, float* __restrict__ out)
{
    if (threadIdx.x != 0 || blockIdx.x != 0) return;
    float loss_node = 0.f;
    for (int c = 0; c < 64; ++c) loss_node += lossc[c];
    float loss_hier = 0.f;
    for (int c = 0; c < 16; ++c) loss_hier += lossc[64 + c];
    out[0] = loss_node / (float)C_CH;
    out[1] = loss_hier / (2.f * (float)C_CH);
}

// ---------------------------------------------------------------------------
extern "C" void kernel_launch(void* const* d_in, const int* in_sizes, int n_in,
                              void* d_out, int out_size, void* d_ws, size_t ws_size,
                              hipStream_t stream)
{
    const float* conv  = (const float*)d_in[0];
    const int*   im2c0 = (const int*)d_in[1];
    const int*   im2c1 = (const int*)d_in[2];
    float* out = (float*)d_out;

    char* ws = (char*)d_ws;
    size_t off = 0;
    auto carve = [&](size_t bytes) -> char* {
        char* p = ws + off;
        off += (bytes + 255) & ~size_t(255);
        return p;
    };
    int*   members  = (int*)  carve(2 * C_CH * sizeof(int));
    int*   rowstart = (int*)  carve(2 * 64 * sizeof(int));
    int*   count    = (int*)  carve(2 * 64 * sizeof(int));
    int*   rowclu   = (int*)  carve(2 * C_CH * sizeof(int));
    int*   jbase    = (int*)  carve(2 * C_CH * 8 * sizeof(int));
    float* P0       = (float*)carve((size_t)NROWS * 64 * sizeof(float));
    float* P1       = (float*)carve((size_t)NROWS * 16 * sizeof(float));
    float* rn       = (float*)carve(2 * NROWS * sizeof(float));
    float* concv    = (float*)carve(2 * 64 * sizeof(float));
    float* lossc    = (float*)carve(2 * 64 * sizeof(float));
    (void)ws_size; (void)in_sizes; (void)n_in; (void)out_size;

    k_build<<<2, 32, 0, stream>>>(im2c0, im2c1, members, rowstart, count, rowclu);
    k_jbase<<<256, 256, 0, stream>>>(members, rowstart, count, rowclu, jbase);

    // level 0: K = 64
    k_gemm<<<NROWS / 16, 128, 0, stream>>>(conv, jbase, rowstart,
                                           P0, rn, 64);
    // level 1: K = 16
    k_gemm<<<NROWS / 16, 128, 0, stream>>>(conv, jbase + C_CH * 8, rowstart + 64,
                                           P1, rn + NROWS, 16);

    k_cluster<<<80, 256, 0, stream>>>(rowstart, count, P0, P1, rn, concv, lossc);
    k_final<<<1, 32, 0, stream>>>(lossc, out);
}